// DSSnetwork_627065225442
// MI455X (gfx1250) — compile-verified
//
#include <hip/hip_runtime.h>
#include <hip/hip_bf16.h>

// Static problem structure (must match setup_inputs): G=64, S=50, N=50, D=128, L=4.
#define DIM   128
#define EPSv  1e-5f

typedef float v2f __attribute__((ext_vector_type(2)));
typedef float v8f __attribute__((ext_vector_type(8)));

// ---------------------------------------------------------------------------
// Fused dual GEMM with f32 WMMA:  out = A0 @ W0 + A1 @ W1 + bias
// A0,A1: M x 128 row-major; W0,W1: 128 x 128 row-major; M must be multiple of 128.
// Block = 256 threads = 8 waves. Block tile = 128 rows x 128 cols.
// Wave w computes rows [blk*128 + 16w, +16) x all 128 cols (8 WMMA tiles, 64 acc VGPRs).
//
// LDS: A tile 128 x 132 (stride-132 padding -> conflict-free b64 fragment reads:
//      dword banks 4r+c cover 64 banks uniquely across a wave), plus the weight
//      matrix pre-swizzled into per-lane WMMA B-fragment layout:
//      bfrag[(kk*8+nt)*32 + lane] = {W[k][n], W[k+1][n]}  (k = 4kk + 2*(lane>>4),
//      n = 16nt + (lane&15)), so each B fetch is one ds_load_b64 whose per-lane
//      dword addresses are {2*lane, 2*lane+1}: all 64 banks exactly once.
// Total LDS = 128*132*4 + 8192*8 = 67584 + 65536 = 133120 B -> 2 workgroups/WGP.
// The two GEMM terms run sequentially through the same LDS with barriers while
// the f32 accumulators stay live.
// ---------------------------------------------------------------------------
#define LDSTRIDE 132

__global__ __launch_bounds__(256) void gemm_dual_wmma(
    const float* __restrict__ A0, const float* __restrict__ A1,
    const float* __restrict__ W0, const float* __restrict__ W1,
    const float* __restrict__ bias, float* __restrict__ out, int M)
{
    __shared__ float atile[128][LDSTRIDE];
    __shared__ v2f   bfrag[32 * 8 * 32];   // [kk][nt][lane] packed K-pairs

    const int tid  = threadIdx.x;
    const int wave = tid >> 5;
    const int lane = tid & 31;
    const int row0 = blockIdx.x * 128;
    if (row0 >= M) return;

    // A fragment geometry (16x4 f32, 2 VGPRs):
    // lanes 0-15 hold K = {k0, k0+1}, lanes 16-31 hold K = {k0+2, k0+3}
    const int mrow = (wave << 4) + (lane & 15);
    const int kb   = (lane >> 4) << 1;     // 0 or 2
    const int ncol = lane & 15;

    v8f acc[8];
    const v8f zeroV = {0.f, 0.f, 0.f, 0.f, 0.f, 0.f, 0.f, 0.f};
#pragma unroll
    for (int nt = 0; nt < 8; ++nt) acc[nt] = zeroV;

    float* bf = (float*)bfrag;

    for (int s = 0; s < 2; ++s) {
        const float* __restrict__ Asrc = s ? A1 : A0;
        const float* __restrict__ W    = s ? W1 : W0;
        if (s) __syncthreads();            // all waves done reading phase-0 LDS

        // Stage A tile: 128 rows x 128 cols, coalesced float4 loads
        for (int it = tid; it < 128 * 32; it += 256) {
            const int r  = it >> 5;
            const int c4 = (it & 31) << 2;
            const float4 v = *reinterpret_cast<const float4*>(
                Asrc + (size_t)(row0 + r) * DIM + c4);
            atile[r][c4 + 0] = v.x;
            atile[r][c4 + 1] = v.y;
            atile[r][c4 + 2] = v.z;
            atile[r][c4 + 3] = v.w;
        }
        // Stage W pre-swizzled into B-fragment layout (read W exactly once/block)
        for (int dIdx = tid; dIdx < 32 * 8 * 32 * 2; dIdx += 256) {
            const int j  = dIdx & 1;
            const int l2 = (dIdx >> 1) & 31;
            const int nt = (dIdx >> 6) & 7;
            const int kk = dIdx >> 9;
            const int k  = (kk << 2) + ((l2 >> 4) << 1) + j;
            const int n  = (nt << 4) + (l2 & 15);
            bf[dIdx] = W[k * DIM + n];
        }
        __syncthreads();

        for (int kk = 0; kk < DIM / 4; ++kk) {
            const int k0 = kk << 2;
            v2f a;
            a.x = atile[mrow][k0 + kb + 0];
            a.y = atile[mrow][k0 + kb + 1];
#pragma unroll
            for (int nt = 0; nt < 8; ++nt) {
                const v2f b = bfrag[((kk << 3) + nt) * 32 + lane];
                // 8-arg pattern: (neg_a, A, neg_b, B, c_mod, C, reuse_a, reuse_b)
                acc[nt] = __builtin_amdgcn_wmma_f32_16x16x4_f32(
                    false, a, false, b, (short)0, acc[nt], false, false);
            }
        }
    }

    // C/D layout: VGPR r -> M = r (lanes 0-15) / r+8 (lanes 16-31), N = lane&15
    const int mbase = (lane >> 4) << 3;
#pragma unroll
    for (int nt = 0; nt < 8; ++nt) {
        const int n    = (nt << 4) + ncol;
        const float bv = bias[n];
#pragma unroll
        for (int r = 0; r < 8; ++r) {
            const int m = mbase + r;
            out[(size_t)(row0 + (wave << 4) + m) * DIM + n] = acc[nt][r] + bv;
        }
    }
}

// ---------------------------------------------------------------------------
// Elementwise / scatter helpers
// ---------------------------------------------------------------------------
__global__ void zero_f32_kernel(float* __restrict__ p, long n)
{
    const long gid = (long)blockIdx.x * blockDim.x + threadIdx.x;
    if (gid < n) p[gid] = 0.f;
}

__global__ void copy4_kernel(const float* __restrict__ src, float* __restrict__ dst, long n4)
{
    const long gid = (long)blockIdx.x * blockDim.x + threadIdx.x;
    if (gid < n4) reinterpret_cast<float4*>(dst)[gid] =
                      reinterpret_cast<const float4*>(src)[gid];
}

// One wave per edge; lane handles 4 columns. agg[dst] += X[src]
__global__ void edge_scatter_kernel(const float* __restrict__ X,
    const int* __restrict__ src, const int* __restrict__ dst,
    float* __restrict__ agg, int E)
{
    const long gid = (long)blockIdx.x * blockDim.x + threadIdx.x;
    const int e = (int)(gid >> 5);
    const int lane = (int)(gid & 31);
    if (e >= E) return;
    const int sv = src[e], dv = dst[e];
    const float4 v = reinterpret_cast<const float4*>(X + (size_t)sv * DIM)[lane];
    float* d = agg + (size_t)dv * DIM + lane * 4;
    atomicAdd(d + 0, v.x); atomicAdd(d + 1, v.y);
    atomicAdd(d + 2, v.z); atomicAdd(d + 3, v.w);
}

// One wave per row; out[idx[v]] += X[v]
__global__ void row_scatter_kernel(const float* __restrict__ X,
    const int* __restrict__ idx, float* __restrict__ out, int rows)
{
    const long gid = (long)blockIdx.x * blockDim.x + threadIdx.x;
    const int v = (int)(gid >> 5);
    const int lane = (int)(gid & 31);
    if (v >= rows) return;
    const int t = idx[v];
    const float4 x4 = reinterpret_cast<const float4*>(X + (size_t)v * DIM)[lane];
    float* d = out + (size_t)t * DIM + lane * 4;
    atomicAdd(d + 0, x4.x); atomicAdd(d + 1, x4.y);
    atomicAdd(d + 2, x4.z); atomicAdd(d + 3, x4.w);
}

__global__ void count_kernel(const int* __restrict__ idx, float* __restrict__ cnt, int rows)
{
    const int v = blockIdx.x * blockDim.x + threadIdx.x;
    if (v < rows) atomicAdd(&cnt[idx[v]], 1.0f);
}

__global__ void scale_rows_kernel(float* __restrict__ X, const float* __restrict__ cnt, int rows)
{
    const long gid = (long)blockIdx.x * blockDim.x + threadIdx.x;
    if (gid >= (long)rows * DIM) return;
    const int r = (int)(gid >> 7);
    X[gid] = X[gid] / fmaxf(cnt[r], 1.0f);
}

// Column sums + sums of squares (for batch-stat BN). Block = 128 threads,
// thread t owns column t; each block handles 128 rows; atomic partial reduce.
__global__ __launch_bounds__(128) void colstats_kernel(
    const float* __restrict__ H, float* __restrict__ stats, int rows)
{
    const int c  = threadIdx.x;
    const int r0 = blockIdx.x * 128;
    const int r1 = min(r0 + 128, rows);
    float s = 0.f, ss = 0.f;
    for (int r = r0; r < r1; ++r) {
        const float v = H[(size_t)r * DIM + c];
        s += v; ss += v * v;
    }
    atomicAdd(&stats[c], s);
    atomicAdd(&stats[DIM + c], ss);
}

// x = relu( BN(h1) + BN(h2)[node_idx] ), fully fused
__global__ void bn_combine_relu_kernel(const float* __restrict__ h1,
    const float* __restrict__ h2, const int* __restrict__ nidx,
    const float* __restrict__ st1, const float* __restrict__ st2,
    const float* __restrict__ g1, const float* __restrict__ be1,
    const float* __restrict__ g2, const float* __restrict__ be2,
    float* __restrict__ xout, int rows, float invM1, float invM2)
{
    const long gid = (long)blockIdx.x * blockDim.x + threadIdx.x;
    if (gid >= (long)rows * DIM) return;
    const int c = (int)(gid & (DIM - 1));
    const int v = (int)(gid >> 7);
    const float mu1  = st1[c] * invM1;
    const float var1 = fmaxf(st1[DIM + c] * invM1 - mu1 * mu1, 0.f);
    const float mu2  = st2[c] * invM2;
    const float var2 = fmaxf(st2[DIM + c] * invM2 - mu2 * mu2, 0.f);
    const float a  = g1[c] * (h1[gid] - mu1) * rsqrtf(var1 + EPSv) + be1[c];
    const float bb = g2[c] * (h2[(size_t)nidx[v] * DIM + c] - mu2) * rsqrtf(var2 + EPSv) + be2[c];
    xout[gid] = fmaxf(a + bb, 0.f);
}

// Offsets: offs[0..G] = cumsum(num_nodes_per_subgraph), offs[G+1..2G+1] = cumsum(num_subgraphs)
__global__ void prefix_kernel(const int* __restrict__ nnps,
    const int* __restrict__ nsub, int* __restrict__ offs, int Gn)
{
    if (blockIdx.x == 0 && threadIdx.x == 0) {
        int a = 0, b2 = 0;
        for (int g = 0; g < Gn; ++g) {
            offs[g] = a; offs[Gn + 1 + g] = b2;
            a += nnps[g]; b2 += nsub[g];
        }
        offs[Gn] = a; offs[2 * Gn + 1] = b2;
    }
}

__global__ void idx_map_kernel(const int* __restrict__ batch,
    const int* __restrict__ sni, const int* __restrict__ sb,
    const int* __restrict__ offs, int Gn,
    int* __restrict__ nidx, int* __restrict__ sidx, int rows)
{
    const int v = blockIdx.x * blockDim.x + threadIdx.x;
    if (v >= rows) return;
    const int g = batch[v];
    nidx[v] = offs[g] + sni[v];
    sidx[v] = offs[Gn + 1 + g] + sb[v];
}

// MLP head: hidden = relu(hg @ fW1 + fb1)   (grid = G blocks, 256 threads)
__global__ __launch_bounds__(256) void fc1_kernel(const float* __restrict__ hg,
    const float* __restrict__ W1, const float* __restrict__ b1f,
    float* __restrict__ hidden)
{
    const int g = blockIdx.x, j = threadIdx.x;
    float s = b1f[j];
    for (int k = 0; k < DIM; ++k) s += hg[g * DIM + k] * W1[k * 256 + j];
    hidden[g * 256 + j] = fmaxf(s, 0.f);
}

__global__ void fc2_kernel(const float* __restrict__ hidden,
    const float* __restrict__ W2, const float* __restrict__ b2f,
    float* __restrict__ out, int Gn)
{
    const int gid = blockIdx.x * blockDim.x + threadIdx.x;
    if (gid >= Gn * 10) return;
    const int g = gid / 10, o = gid % 10;
    float s = b2f[o];
    for (int k = 0; k < 256; ++k) s += hidden[g * 256 + k] * W2[k * 10 + o];
    out[gid] = s;
}

// ---------------------------------------------------------------------------
// Host orchestration
// ---------------------------------------------------------------------------
extern "C" void kernel_launch(void* const* d_in, const int* in_sizes, int n_in,
                              void* d_out, int out_size, void* d_ws, size_t ws_size,
                              hipStream_t stream)
{
    const float* x_in   = (const float*)d_in[0];
    const float* Wr     = (const float*)d_in[1];
    const float* Wn     = (const float*)d_in[2];
    const float* bconv  = (const float*)d_in[3];
    const float* bng    = (const float*)d_in[4];
    const float* bnb    = (const float*)d_in[5];
    const float* Wsr    = (const float*)d_in[6];
    const float* Wsn    = (const float*)d_in[7];
    const float* bsconv = (const float*)d_in[8];
    const float* bnsg   = (const float*)d_in[9];
    const float* bnsb   = (const float*)d_in[10];
    const float* fW1    = (const float*)d_in[11];
    const float* fb1    = (const float*)d_in[12];
    const float* fW2    = (const float*)d_in[13];
    const float* fb2    = (const float*)d_in[14];
    const int*   eidx   = (const int*)d_in[15];
    const int*   oeidx  = (const int*)d_in[16];
    const int*   batch  = (const int*)d_in[17];
    const int*   sbat   = (const int*)d_in[18];
    const int*   sni    = (const int*)d_in[19];
    const int*   nsub   = (const int*)d_in[20];
    const int*   nnps   = (const int*)d_in[21];
    const int*   sib    = (const int*)d_in[22];

    const int TN    = in_sizes[17];          // 160000
    const int Gn    = in_sizes[20];          // 64
    const int NSUBG = in_sizes[22];          // 3200
    const int Nn    = TN / NSUBG;            // 50
    const int NORIG = Gn * Nn;               // 3200
    const int Lg    = in_sizes[3] / DIM;     // 4
    const int E     = in_sizes[15] / 2;      // 1,280,000
    const int Eo    = in_sizes[16] / 2;      // 25,600

    const int* esrc = eidx;         const int* edst = eidx + E;
    const int* osrc = oeidx;        const int* odst = oeidx + Eo;

    // ---- workspace layout ----
    char* ws = (char*)d_ws;
    size_t off = 0;
    auto walloc = [&](size_t bytes) -> void* {
        void* p = ws + off;
        off += (bytes + 255) & ~(size_t)255;
        return p;
    };
    float* xbuf   = (float*)walloc((size_t)TN * DIM * 4);
    float* agg    = (float*)walloc((size_t)TN * DIM * 4);
    float* h1     = (float*)walloc((size_t)TN * DIM * 4);
    float* xsum   = (float*)walloc((size_t)NORIG * DIM * 4);
    float* oagg   = (float*)walloc((size_t)NORIG * DIM * 4);
    float* h2     = (float*)walloc((size_t)NORIG * DIM * 4);
    float* hsub   = (float*)walloc((size_t)NSUBG * DIM * 4);
    float* hgraph = (float*)walloc((size_t)Gn * DIM * 4);
    float* hidden = (float*)walloc((size_t)Gn * 256 * 4);
    float* stats1 = (float*)walloc(2 * DIM * 4);
    float* stats2 = (float*)walloc(2 * DIM * 4);
    float* cntO   = (float*)walloc((size_t)NORIG * 4);
    float* cntS   = (float*)walloc((size_t)NSUBG * 4);
    float* cntG   = (float*)walloc((size_t)Gn * 4);
    int*   nidx   = (int*)walloc((size_t)TN * 4);
    int*   sidx   = (int*)walloc((size_t)TN * 4);
    int*   offs   = (int*)walloc((size_t)(2 * Gn + 2) * 4);
    if (off > ws_size) return;  // workspace too small: deterministic no-op

    auto zero = [&](float* p, long n) {
        zero_f32_kernel<<<(int)((n + 255) / 256), 256, 0, stream>>>(p, n);
    };

    // ---- precompute index maps ----
    prefix_kernel<<<1, 1, 0, stream>>>(nnps, nsub, offs, Gn);
    idx_map_kernel<<<(TN + 255) / 256, 256, 0, stream>>>(batch, sni, sbat, offs, Gn,
                                                         nidx, sidx, TN);
    {
        const long n4 = (long)TN * (DIM / 4);
        copy4_kernel<<<(int)((n4 + 255) / 256), 256, 0, stream>>>(x_in, xbuf, n4);
    }

    const float invTN = 1.0f / (float)TN;
    const float invNO = 1.0f / (float)NORIG;

    for (int i = 0; i < Lg; ++i) {
        const size_t wo = (size_t)i * DIM * DIM;
        const size_t vo = (size_t)i * DIM;

        // (a) node-level graph conv on subgraph copies
        zero(agg, (long)TN * DIM);
        edge_scatter_kernel<<<(int)(((long)E * 32 + 255) / 256), 256, 0, stream>>>(
            xbuf, esrc, edst, agg, E);
        gemm_dual_wmma<<<TN / 128, 256, 0, stream>>>(
            xbuf, agg, Wr + wo, Wn + wo, bconv + vo, h1, TN);
        zero(stats1, 2 * DIM);
        colstats_kernel<<<(TN + 127) / 128, 128, 0, stream>>>(h1, stats1, TN);

        // (b) subgraph-info path: scatter-mean to original nodes, conv, stats
        zero(xsum, (long)NORIG * DIM);
        zero(cntO, NORIG);
        row_scatter_kernel<<<(int)(((long)TN * 32 + 255) / 256), 256, 0, stream>>>(
            xbuf, nidx, xsum, TN);
        count_kernel<<<(TN + 255) / 256, 256, 0, stream>>>(nidx, cntO, TN);
        scale_rows_kernel<<<(int)(((long)NORIG * DIM + 255) / 256), 256, 0, stream>>>(
            xsum, cntO, NORIG);
        zero(oagg, (long)NORIG * DIM);
        edge_scatter_kernel<<<(int)(((long)Eo * 32 + 255) / 256), 256, 0, stream>>>(
            xsum, osrc, odst, oagg, Eo);
        gemm_dual_wmma<<<NORIG / 128, 256, 0, stream>>>(
            xsum, oagg, Wsr + wo, Wsn + wo, bsconv + vo, h2, NORIG);
        zero(stats2, 2 * DIM);
        colstats_kernel<<<(NORIG + 127) / 128, 128, 0, stream>>>(h2, stats2, NORIG);

        // (c) fused BN + BN-gather + add + relu, written in place over x
        bn_combine_relu_kernel<<<(int)(((long)TN * DIM + 255) / 256), 256, 0, stream>>>(
            h1, h2, nidx, stats1, stats2,
            bng + vo, bnb + vo, bnsg + vo, bnsb + vo,
            xbuf, TN, invTN, invNO);
    }

    // ---- hierarchical mean pooling ----
    zero(hsub, (long)NSUBG * DIM);
    zero(cntS, NSUBG);
    row_scatter_kernel<<<(int)(((long)TN * 32 + 255) / 256), 256, 0, stream>>>(
        xbuf, sidx, hsub, TN);
    count_kernel<<<(TN + 255) / 256, 256, 0, stream>>>(sidx, cntS, TN);
    scale_rows_kernel<<<(int)(((long)NSUBG * DIM + 255) / 256), 256, 0, stream>>>(
        hsub, cntS, NSUBG);

    zero(hgraph, (long)Gn * DIM);
    zero(cntG, Gn);
    row_scatter_kernel<<<(int)(((long)NSUBG * 32 + 255) / 256), 256, 0, stream>>>(
        hsub, sib, hgraph, NSUBG);
    count_kernel<<<(NSUBG + 255) / 256, 256, 0, stream>>>(sib, cntG, NSUBG);
    scale_rows_kernel<<<(int)(((long)Gn * DIM + 255) / 256), 256, 0, stream>>>(
        hgraph, cntG, Gn);

    // ---- MLP head ----
    fc1_kernel<<<Gn, 256, 0, stream>>>(hgraph, fW1, fb1, hidden);
    fc2_kernel<<<(Gn * 10 + 63) / 64, 64, 0, stream>>>(hidden, fW2, fb2,
                                                       (float*)d_out, Gn);
}